// SpanNER_18433999635248
// MI455X (gfx1250) — compile-verified
//
#include <hip/hip_runtime.h>
#include <hip/hip_bf16.h>

// ---------------------------------------------------------------------------
// SpanNER fused kernel for MI455X (gfx1250, wave32).
//
// Only 189 of the 2080 span slots per sequence are non-pad; pad rows have
// rep == b_span, i.e. constant logit / constant 128-vector projection.
// Kernel 1 (fused): per (batch, 16-row M-tile): gathered-A GEMM (K=2304) ->
// rep tile in LDS -> logits (dot+shuffle) -> GEMM (K=768) -> relu h tile in
// LDS -> GEMM (K=768, N=128) -> projections. All matmuls via
// v_wmma_f32_16x16x4_f32 (fp32-exact matrix pipe).
// Kernel 2: broadcast the pad-constant row (computed by kernel 1 via the
// zero-masked rows 189..191) into slots 192..2079 of every batch.
// ---------------------------------------------------------------------------

typedef __attribute__((ext_vector_type(2))) float v2f;
typedef __attribute__((ext_vector_type(8))) float v8f;

#define BATCH       32
#define SEQ         64
#define DIM         768
#define PROJ_DIM    128
#define K_ACT       189      // real spans per sequence
#define M_TILES     12       // 12 * 16 = 192 >= 189
#define M_PAD       (M_TILES * 16)
#define TOTAL_SPANS 2080     // 64*65/2
#define NWAVES      16
#define LDS_PITCH   (DIM + 4)   // stride%64 == 4 floats -> conflict-free rows

__device__ __forceinline__ v8f wmma4(v2f a, v2f b, v8f c) {
  // D = A(16x4,f32) * B(4x16,f32) + C(16x16,f32)
  return __builtin_amdgcn_wmma_f32_16x16x4_f32(
      /*neg_a=*/false, a, /*neg_b=*/false, b,
      /*c_mod=*/(short)0, c, /*reuse_a=*/false, /*reuse_b=*/false);
}

// One 768-deep K-panel: A rows come from `ap` (per-lane row base, global or
// LDS; `mask` zeroes pad rows without breaking EXEC), B is row-major `wp`
// with leading dim DIM; wave owns columns [n, n+48) = 3 n-tiles.
__device__ __forceinline__ void panel768(const float* __restrict__ ap, float mask,
                                         const float* __restrict__ wp, int n,
                                         int lhalf, int lrow,
                                         v8f& c0, v8f& c1, v8f& c2) {
  for (int k0 = 0; k0 < DIM; k0 += 4) {
    const int e = k0 + 2 * lhalf;          // A: VGPR0=K(e), VGPR1=K(e+1)
    v2f a;
    a.x = ap[e] * mask;
    a.y = ap[e + 1] * mask;
    const float* wb = wp + (size_t)e * DIM + n + lrow;  // B row e, col n+lrow
    v2f b0, b1, b2;
    b0.x = wb[0];       b0.y = wb[DIM];          // B: VGPR0=row e, VGPR1=row e+1
    b1.x = wb[16];      b1.y = wb[DIM + 16];
    b2.x = wb[32];      b2.y = wb[DIM + 32];
    c0 = wmma4(a, b0, c0);
    c1 = wmma4(a, b1, c1);
    c2 = wmma4(a, b2, c2);
  }
}

// Store one 16x16 f32 D-tile into an LDS row-major tile with bias (+optional
// relu). D layout: VGPR v -> row v (lanes 0-15) / row v+8 (lanes 16-31).
__device__ __forceinline__ void store_tile(float* __restrict__ dst, const v8f& c,
                                           int ncol0, const float* __restrict__ bias,
                                           int lhalf, int lrow, bool relu) {
  const int col = ncol0 + lrow;
  const float bv = bias[col];
#pragma unroll
  for (int v = 0; v < 8; ++v) {
    const int row = v + 8 * lhalf;
    float x = c[v] + bv;
    if (relu) x = fmaxf(x, 0.0f);
    dst[row * LDS_PITCH + col] = x;
  }
}

__global__ __launch_bounds__(NWAVES * 32, 1)
void spanner_fused(const float* __restrict__ seq,       // (B,S,D)
                   const float* __restrict__ width_emb, // (3,D)
                   const float* __restrict__ w_span,    // (3D,D)
                   const float* __restrict__ b_span,    // (D)
                   const float* __restrict__ w_cls,     // (D)
                   const float* __restrict__ b_cls,     // (1)
                   const float* __restrict__ w_p1,      // (D,D)
                   const float* __restrict__ b_p1,      // (D)
                   const float* __restrict__ w_p2,      // (D,PROJ)
                   const float* __restrict__ b_p2,      // (PROJ)
                   float* __restrict__ logits,          // (B,2080)
                   float* __restrict__ proj)            // (B,2080,PROJ)
{
  extern __shared__ float lds[];
  float* repS = lds;                       // [16][LDS_PITCH]
  float* hS   = lds + 16 * LDS_PITCH;      // [16][LDS_PITCH]

  const int b    = blockIdx.x / M_TILES;
  const int mt   = blockIdx.x % M_TILES;
  const int wave = threadIdx.x >> 5;
  const int lane = threadIdx.x & 31;
  const int lrow = lane & 15;              // A: row in tile / B: col in n-tile
  const int lhalf = lane >> 4;

  // ---- per-lane span gather setup (row m = lrow of this M-tile) ----
  const int t = mt * 16 + lrow;
  int si, wid;
  if (t < 186)      { si = t / 3; wid = t - si * 3; }
  else if (t < 188) { si = 62;    wid = t - 186;    }
  else              { si = 63;    wid = 0;          }
  const int ei = si + wid;
  const float mask = (t < K_ACT) ? 1.0f : 0.0f;    // zero rows 189..191 == pad
  const float* pS = seq + ((size_t)b * SEQ + si) * DIM;
  const float* pE = seq + ((size_t)b * SEQ + ei) * DIM;
  const float* pW = width_emb + (size_t)wid * DIM;

  const int n0 = wave * 48;                // 16 waves cover N = 768

  // ---- GEMM1: rep = [starts|ends|width] @ w_span + b_span  (K = 2304) ----
  v8f c0 = {}, c1 = {}, c2 = {};
  panel768(pS, mask, w_span,                      n0, lhalf, lrow, c0, c1, c2);
  panel768(pE, mask, w_span + (size_t)DIM * DIM,  n0, lhalf, lrow, c0, c1, c2);
  panel768(pW, mask, w_span + (size_t)2*DIM*DIM,  n0, lhalf, lrow, c0, c1, c2);
  store_tile(repS, c0, n0,      b_span, lhalf, lrow, false);
  store_tile(repS, c1, n0 + 16, b_span, lhalf, lrow, false);
  store_tile(repS, c2, n0 + 32, b_span, lhalf, lrow, false);
  __syncthreads();

  // ---- logits: wave r reduces row r of repS against w_cls ----
  {
    float sum = 0.0f;
    const float* rr = repS + wave * LDS_PITCH;
    for (int e = lane; e < DIM; e += 32) sum += rr[e] * w_cls[e];
#pragma unroll
    for (int off = 16; off > 0; off >>= 1) sum += __shfl_xor(sum, off, 32);
    if (lane == 0)
      logits[(size_t)b * TOTAL_SPANS + mt * 16 + wave] = sum + b_cls[0];
  }

  // ---- GEMM2: h = relu(rep @ w_p1 + b_p1)  (K = 768) ----
  c0 = (v8f){}; c1 = (v8f){}; c2 = (v8f){};
  panel768(repS + lrow * LDS_PITCH, 1.0f, w_p1, n0, lhalf, lrow, c0, c1, c2);
  store_tile(hS, c0, n0,      b_p1, lhalf, lrow, true);
  store_tile(hS, c1, n0 + 16, b_p1, lhalf, lrow, true);
  store_tile(hS, c2, n0 + 32, b_p1, lhalf, lrow, true);
  __syncthreads();

  // ---- GEMM3: proj = h @ w_p2 + b_p2  (K = 768, N = 128; waves 0..7) ----
  if (wave < 8) {
    v8f cp = {};
    const int np = wave * 16;
    const float* hp = hS + lrow * LDS_PITCH;
    for (int k0 = 0; k0 < DIM; k0 += 4) {
      const int e = k0 + 2 * lhalf;
      v2f a; a.x = hp[e]; a.y = hp[e + 1];
      const float* wb = w_p2 + (size_t)e * PROJ_DIM + np + lrow;
      v2f bb; bb.x = wb[0]; bb.y = wb[PROJ_DIM];
      cp = wmma4(a, bb, cp);
    }
    const float bv = b_p2[np + lrow];
#pragma unroll
    for (int v = 0; v < 8; ++v) {
      const int row = mt * 16 + v + 8 * lhalf;
      proj[((size_t)b * TOTAL_SPANS + row) * PROJ_DIM + np + lrow] = cp[v] + bv;
    }
  }
}

// Broadcast pad-constant outputs (row t=189 of batch 0 is a pad row already
// computed by the fused kernel) into slots t in [192, 2080) of every batch.
__global__ __launch_bounds__(128)
void spanner_pad_fill(float* __restrict__ out) {
  float* logits = out;
  float* proj   = out + (size_t)BATCH * TOTAL_SPANS;
  const float padLogit = logits[K_ACT];                         // b=0, t=189
  const float padProj  = proj[(size_t)K_ACT * PROJ_DIM + threadIdx.x];
  const int padPerB = TOTAL_SPANS - M_PAD;                      // 1888
  const int nRows = BATCH * padPerB;
  for (int r = blockIdx.x; r < nRows; r += gridDim.x) {
    const int b = r / padPerB;
    const int t = M_PAD + (r - b * padPerB);
    if (threadIdx.x == 0) logits[(size_t)b * TOTAL_SPANS + t] = padLogit;
    proj[((size_t)b * TOTAL_SPANS + t) * PROJ_DIM + threadIdx.x] = padProj;
  }
}

extern "C" void kernel_launch(void* const* d_in, const int* in_sizes, int n_in,
                              void* d_out, int out_size, void* d_ws, size_t ws_size,
                              hipStream_t stream) {
  (void)in_sizes; (void)n_in; (void)out_size; (void)d_ws; (void)ws_size;
  const float* seq  = (const float*)d_in[0];
  const float* wemb = (const float*)d_in[1];
  const float* wsp  = (const float*)d_in[2];
  const float* bsp  = (const float*)d_in[3];
  const float* wcls = (const float*)d_in[4];
  const float* bcls = (const float*)d_in[5];
  const float* wp1  = (const float*)d_in[6];
  const float* bp1  = (const float*)d_in[7];
  const float* wp2  = (const float*)d_in[8];
  const float* bp2  = (const float*)d_in[9];

  float* out    = (float*)d_out;
  float* logits = out;
  float* proj   = out + (size_t)BATCH * TOTAL_SPANS;

  const size_t smem = (size_t)2 * 16 * LDS_PITCH * sizeof(float);  // ~96.5 KB
  spanner_fused<<<dim3(BATCH * M_TILES), dim3(NWAVES * 32), smem, stream>>>(
      seq, wemb, wsp, bsp, wcls, bcls, wp1, bp1, wp2, bp2, logits, proj);

  spanner_pad_fill<<<dim3(2048), dim3(PROJ_DIM), 0, stream>>>(out);
}